// CapsuleLayer_17489106829902
// MI455X (gfx1250) — compile-verified
//
#include <hip/hip_runtime.h>
#include <math.h>

// Problem constants (match reference)
#define B_    32      // batch
#define NIN   2048    // input capsules
#define DD    32      // input capsule dim (contraction)
#define JJ    64      // output capsules
#define KK    32      // output capsule dim
// Tiling
#define BBLK  16      // batch rows per workgroup (WMMA M)
#define NT    32      // input capsules per workgroup tile
#define TT    (NIN / NT)       // 64 n-tiles
#define WAVES 16
#define JW    (JJ / WAVES)     // 4 output capsules per wave
#define THREADS (WAVES * 32)   // 512 (wave32)

typedef float v2f __attribute__((ext_vector_type(2)));
typedef float v8f __attribute__((ext_vector_type(8)));

__device__ __forceinline__ v8f wmma_f32_16x16x4(v2f a, v2f b, v8f c) {
  // D = A(16x4 f32) * B(4x16 f32) + C(16x16 f32), wave32 WMMA
  return __builtin_amdgcn_wmma_f32_16x16x4_f32(false, a, false, b, (short)0, c,
                                               false, false);
}

// One fused routing pass over an n-tile. Computes u_hat tiles via WMMA,
// (optionally) logits = u_hat . vsum, workgroup-local softmax over J,
// and the c-weighted partial sum s_partial[b,j,k] for this tile.
template <bool HAS_V>
__global__ __launch_bounds__(THREADS, 1)
void caps_pass(const float* __restrict__ x, const float* __restrict__ W,
               const float* __restrict__ vsum, float* __restrict__ sPartial)
{
  __shared__ float logits[BBLK][JJ];   // per-(b,n) logits, then c after softmax
  __shared__ float red[BBLK][16];
  __shared__ float rowMax[BBLK];
  __shared__ float rowInv[BBLK];

  const int tid  = threadIdx.x;
  const int lane = tid & 31;
  const int wave = tid >> 5;
  const int ll   = lane & 15;   // lane-in-half
  const int lh   = lane >> 4;   // half select
  const int t    = blockIdx.x;  // n tile
  const int bb   = blockIdx.y;  // batch block (0/1)
  const int b0   = bb * BBLK;
  const int n0   = t * NT;
  const int j0   = wave * JW;

  const v8f vzero = {};
  v8f s_acc[JW][2];   // s partial, D-frag layout: [r]->b=r+8*lh, k=16*h+ll
  v8f u[JW][2];       // u_hat fragments (HAS_V only)
  v8f vf[JW][2];      // vsum fragments  (HAS_V only)
#pragma unroll
  for (int jj = 0; jj < JW; ++jj)
#pragma unroll
    for (int h = 0; h < 2; ++h) s_acc[jj][h] = vzero;

  if (HAS_V) {
#pragma unroll
    for (int jj = 0; jj < JW; ++jj)
#pragma unroll
      for (int h = 0; h < 2; ++h) {
        v8f tmp = vzero;
#pragma unroll
        for (int r = 0; r < 8; ++r)
          tmp[r] = vsum[((size_t)(b0 + r + 8 * lh) * JJ + (j0 + jj)) * KK +
                        16 * h + ll];
        vf[jj][h] = tmp;
      }
  }

  const float* xrow = x + (size_t)(b0 + ll) * NIN * DD;

  for (int n = n0; n < n0 + NT; ++n) {
    // --- A fragments: a[s] = x[b0+ll, n, 4s+2lh .. +1]  (16x4 tile per step)
    v2f a[8];
    const float* xp = xrow + (size_t)n * DD + 2 * lh;
#pragma unroll
    for (int s = 0; s < 8; ++s) a[s] = *(const v2f*)(xp + 4 * s);

    // --- u_hat[b,k] = sum_d x[b,n,d] * W[n,j,d,k]  via 16 WMMAs per j
#pragma unroll
    for (int jj = 0; jj < JW; ++jj) {
      const float* wp = W + ((size_t)n * JJ + (j0 + jj)) * DD * KK +
                        (size_t)(2 * lh) * KK + ll;
#pragma unroll
      for (int h = 0; h < 2; ++h) {
        v8f acc = HAS_V ? vzero : s_acc[jj][h];  // !HAS_V: chain C across n
        const float* wph = wp + 16 * h;
#pragma unroll
        for (int s = 0; s < 8; ++s) {
          v2f bf;
          bf.x = wph[(size_t)(4 * s) * KK];
          bf.y = wph[(size_t)(4 * s + 1) * KK];
          acc = wmma_f32_16x16x4(a[s], bf, acc);
        }
        if (HAS_V) u[jj][h] = acc; else s_acc[jj][h] = acc;
      }
    }

    if (HAS_V) {
      // --- logits[b][j] = sum_k u_hat * vsum (cross-lane reduce over k)
#pragma unroll
      for (int jj = 0; jj < JW; ++jj) {
        v8f dot = u[jj][0] * vf[jj][0] + u[jj][1] * vf[jj][1];
#pragma unroll
        for (int r = 0; r < 8; ++r) {
          float p = dot[r];
          p += __shfl_xor(p, 1, 32);
          p += __shfl_xor(p, 2, 32);
          p += __shfl_xor(p, 4, 32);
          p += __shfl_xor(p, 8, 32);           // stays within 16-lane half
          if (ll == 0) logits[r + 8 * lh][j0 + jj] = p;
        }
      }
      __syncthreads();
      // --- softmax over j for each of 16 b rows (all J local to this WG)
      if (tid < 256) {
        int b = tid >> 4, q = tid & 15;
        float m = -1e30f;
        for (int w = 0; w < 4; ++w) m = fmaxf(m, logits[b][q + 16 * w]);
        red[b][q] = m;
      }
      __syncthreads();
      if (tid < BBLK) {
        float m = -1e30f;
        for (int q = 0; q < 16; ++q) m = fmaxf(m, red[tid][q]);
        rowMax[tid] = m;
      }
      __syncthreads();
      if (tid < 256) {
        int b = tid >> 4, q = tid & 15;
        float m = rowMax[b], sum = 0.f;
        for (int w = 0; w < 4; ++w) {
          float e = __expf(logits[b][q + 16 * w] - m);
          logits[b][q + 16 * w] = e;
          sum += e;
        }
        red[b][q] = sum;
      }
      __syncthreads();
      if (tid < BBLK) {
        float sum = 0.f;
        for (int q = 0; q < 16; ++q) sum += red[tid][q];
        rowInv[tid] = 1.0f / sum;
      }
      __syncthreads();
      // --- s += c * u_hat  (c per D-frag element, varies with r -> b)
#pragma unroll
      for (int jj = 0; jj < JW; ++jj) {
        v8f c = vzero;
#pragma unroll
        for (int r = 0; r < 8; ++r) {
          int b = r + 8 * lh;
          c[r] = logits[b][j0 + jj] * rowInv[b];
        }
        s_acc[jj][0] += c * u[jj][0];
        s_acc[jj][1] += c * u[jj][1];
      }
      __syncthreads();  // logits reused next n
    }
  }

  // --- write per-tile partial s (deterministically reduced later)
  float* sp = sPartial + (size_t)(bb * TT + t) * BBLK * JJ * KK;
#pragma unroll
  for (int jj = 0; jj < JW; ++jj)
#pragma unroll
    for (int h = 0; h < 2; ++h)
#pragma unroll
      for (int r = 0; r < 8; ++r)
        sp[((size_t)(r + 8 * lh) * JJ + (j0 + jj)) * KK + 16 * h + ll] =
            s_acc[jj][h][r];
}

// Fixed-order reduction over n-tiles + squash. One 32-lane wave per (b,j).
__global__ __launch_bounds__(KK)
void caps_reduce_squash(const float* __restrict__ sPartial,
                        float* __restrict__ vsum, float* __restrict__ vout,
                        int accumulate, float cscale)
{
  const int bj = blockIdx.x;          // b*JJ + j
  const int b = bj / JJ, j = bj % JJ;
  const int bb = b >> 4, bl = b & 15;
  const int k = threadIdx.x;

  float s = 0.f;
  for (int t = 0; t < TT; ++t)        // fixed order => deterministic
    s += sPartial[(((size_t)(bb * TT + t) * BBLK + bl) * JJ + j) * KK + k];
  s *= cscale;

  float s2 = s * s;
  s2 += __shfl_xor(s2, 1, 32);
  s2 += __shfl_xor(s2, 2, 32);
  s2 += __shfl_xor(s2, 4, 32);
  s2 += __shfl_xor(s2, 8, 32);
  s2 += __shfl_xor(s2, 16, 32);
  float scale = s2 / ((1.0f + s2) * sqrtf(s2 + 1e-8f));
  float v = scale * s;

  size_t idx = (size_t)bj * KK + k;
  if (accumulate) vsum[idx] += v; else vsum[idx] = v;
  if (vout) vout[idx] = v;
}

extern "C" void kernel_launch(void* const* d_in, const int* in_sizes, int n_in,
                              void* d_out, int out_size, void* d_ws,
                              size_t ws_size, hipStream_t stream)
{
  (void)in_sizes; (void)n_in; (void)out_size; (void)ws_size;
  const float* x = (const float*)d_in[0];   // (B, NIN, DD)
  const float* W = (const float*)d_in[1];   // (NIN, JJ, DD, KK) row-major
  float* out = (float*)d_out;               // (B, JJ, KK)

  // Workspace: 16 MiB partials + 256 KiB vsum (requires ws_size >= ~17 MiB)
  float* sPartial = (float*)d_ws;                               // 2*TT*BBLK*JJ*KK
  float* vsum = sPartial + (size_t)2 * TT * BBLK * JJ * KK;

  dim3 pgrid(TT, 2), pblock(THREADS);
  dim3 rgrid(B_ * JJ), rblock(KK);

  // iter 0: uniform c = 1/J, s0 accumulated directly in WMMA accumulators
  caps_pass<false><<<pgrid, pblock, 0, stream>>>(x, W, nullptr, sPartial);
  caps_reduce_squash<<<rgrid, rblock, 0, stream>>>(sPartial, vsum, nullptr, 0,
                                                   1.0f / (float)JJ);
  // iter 1: logits = u_hat . v0  (b0 == 0, so vsum == v0)
  caps_pass<true><<<pgrid, pblock, 0, stream>>>(x, W, vsum, sPartial);
  caps_reduce_squash<<<rgrid, rblock, 0, stream>>>(sPartial, vsum, nullptr, 1,
                                                   1.0f);
  // iter 2: logits = u_hat . (v0 + v1)
  caps_pass<true><<<pgrid, pblock, 0, stream>>>(x, W, vsum, sPartial);
  caps_reduce_squash<<<rgrid, rblock, 0, stream>>>(sPartial, vsum, out, 1,
                                                   1.0f);
}